// GATMultiDiscrete_28552942584390
// MI455X (gfx1250) — compile-verified
//
#include <hip/hip_runtime.h>

// ---------------- problem constants (match reference) ----------------
#define NNODES 50000
#define NEDGES 800000
#define ETOT   (NEDGES + NNODES)   // edges + self loops
#define NGRAPH 64
#define NHEAD  4
#define NCH    64                  // C
#define HC     256                 // H*C
#define NIN    260

typedef __attribute__((ext_vector_type(2))) float v2f;
typedef __attribute__((ext_vector_type(8))) float v8f;

__device__ __forceinline__ float selu_f(float x) {
  const float scale = 1.0507009873554805f;
  const float alpha = 1.6732632423543772f;
  return scale * (x > 0.f ? x : alpha * (__expf(x) - 1.f));
}

__device__ __forceinline__ void atomicMaxF(float* addr, float val) {
  unsigned int* ua = (unsigned int*)addr;
  unsigned int old = *ua;
  while (__uint_as_float(old) < val) {
    unsigned int assumed = old;
    old = atomicCAS(ua, assumed, __float_as_uint(val));
    if (old == assumed) break;
  }
}

// ---------------------------------------------------------------------
// WMMA f32 GEMM: C[M,NCOLS] = act( (A[M,K] + abias) @ B[K,NCOLS] + cbias )
// All shape/leading-dim/epilogue parameters are template constants so the
// inner loop is branch-free: one b64 A-fragment load + NT*2 B loads +
// NT independent V_WMMA_F32_16X16X4_F32 per K-step.
// Wave computes a 16 x (16*NT) output strip (NT accumulators).
// Requires M%16==0, NCOLS%(16*NT)==0, K%4==0.
// A-frag layout (ISA 7.12.2): lanes0-15 -> M=lane, K=k..k+1 in v0/v1;
// lanes16-31 -> K=k+2..k+3. C/D: VGPR r -> row r (lanes0-15) / r+8 (16-31).
// ---------------------------------------------------------------------
template <int K, int LDA, int LDB, int NCOLS, int NT,
          bool ABIAS, bool CBIAS, bool ACT>
__global__ void wmma_gemm_t(const float* __restrict__ A,
                            const float* __restrict__ abias,
                            const float* __restrict__ B,
                            float* __restrict__ C,
                            const float* __restrict__ cbias, int M) {
  const int lane = threadIdx.x & 31;
  const int wave = threadIdx.x >> 5;
  const int ntn = NCOLS / (16 * NT);          // N-strips per row of tiles
  const int ntiles = (M >> 4) * ntn;
  const int tile = blockIdx.x * 8 + wave;
  if (tile >= ntiles) return;                 // wave-uniform: EXEC stays all-1s
  const int tn = tile % ntn;
  const int tm = tile / ntn;
  const int half = lane >> 4;
  const int l = lane & 15;
  const float* __restrict__ Arow = A + (size_t)(tm * 16 + l) * LDA;
  const int col0 = tn * (16 * NT) + l;

  v8f c[NT];
#pragma unroll
  for (int t = 0; t < NT; ++t) c[t] = (v8f){};

  for (int k = 0; k < K; k += 4) {
    const int ka = k + half * 2;
    v2f a;
    if (ABIAS) {
      a.x = Arow[ka] + abias[ka];
      a.y = Arow[ka + 1] + abias[ka + 1];
    } else {
      a.x = Arow[ka];
      a.y = Arow[ka + 1];
    }
#pragma unroll
    for (int t = 0; t < NT; ++t) {
      v2f b;
      b.x = B[ka * LDB + col0 + 16 * t];
      b.y = B[(ka + 1) * LDB + col0 + 16 * t];
      c[t] = __builtin_amdgcn_wmma_f32_16x16x4_f32(false, a, false, b,
                                                   (short)0, c[t], false, false);
    }
  }

#pragma unroll
  for (int t = 0; t < NT; ++t) {
    const int col = col0 + 16 * t;
    const float cb = CBIAS ? cbias[col] : 0.f;
#pragma unroll
    for (int r = 0; r < 8; ++r) {
      const int row = tm * 16 + r + half * 8;
      float v = c[t][r] + cb;
      if (ACT) v = selu_f(v);
      C[(size_t)row * NCOLS + col] = v;
    }
  }
}

// es[n,h] = sum_c h[n,h*64+c]*a_s[h,c]; ed likewise. Wave per (node,head).
__global__ void attn_scores(const float* __restrict__ h,
                            const float* __restrict__ a_s,
                            const float* __restrict__ a_d,
                            float* __restrict__ es, float* __restrict__ ed) {
  const int lane = threadIdx.x & 31;
  const int id = blockIdx.x * 8 + (threadIdx.x >> 5);
  if (id >= NNODES * NHEAD) return;
  const int node = id >> 2, head = id & 3;
  const float* hr = h + (size_t)node * HC + head * NCH;
  const float* as = a_s + head * NCH;
  const float* ad = a_d + head * NCH;
  const float h0 = hr[lane], h1 = hr[lane + 32];
  float vs = h0 * as[lane] + h1 * as[lane + 32];
  float vd = h0 * ad[lane] + h1 * ad[lane + 32];
#pragma unroll
  for (int o = 16; o > 0; o >>= 1) {
    vs += __shfl_xor(vs, o, 32);
    vd += __shfl_xor(vd, o, 32);
  }
  if (lane == 0) { es[id] = vs; ed[id] = vd; }
}

__global__ void init_layer(float* __restrict__ agg, float* __restrict__ m,
                           float* __restrict__ den) {
  const int idx = blockIdx.x * blockDim.x + threadIdx.x;
  if (idx < NNODES * HC) agg[idx] = 0.f;
  if (idx < NNODES * NHEAD) { m[idx] = -__builtin_inff(); den[idx] = 0.f; }
}

// pass 1: e = leaky_relu(es[src]+ed[dst]); m[dst] = max
__global__ void edge_scores(const int* __restrict__ ei,
                            const float* __restrict__ es,
                            const float* __restrict__ ed,
                            float* __restrict__ ebuf, float* __restrict__ m) {
  const int i = blockIdx.x * blockDim.x + threadIdx.x;
  if (i >= ETOT) return;
  const int src = (i < NEDGES) ? ei[i] : (i - NEDGES);
  const int dst = (i < NEDGES) ? ei[NEDGES + i] : (i - NEDGES);
#pragma unroll
  for (int hh = 0; hh < NHEAD; ++hh) {
    float e = es[src * 4 + hh] + ed[dst * 4 + hh];
    e = (e > 0.f) ? e : 0.2f * e;
    ebuf[(size_t)i * 4 + hh] = e;
    atomicMaxF(&m[dst * 4 + hh], e);
  }
}

// pass 2: ex = exp(e - m[dst]); denom[dst] += ex (ebuf overwritten with ex)
__global__ void edge_softmax_num(const int* __restrict__ ei,
                                 const float* __restrict__ m,
                                 float* __restrict__ ebuf,
                                 float* __restrict__ den) {
  const int i = blockIdx.x * blockDim.x + threadIdx.x;
  if (i >= ETOT) return;
  const int dst = (i < NEDGES) ? ei[NEDGES + i] : (i - NEDGES);
#pragma unroll
  for (int hh = 0; hh < NHEAD; ++hh) {
    const float ex = __expf(ebuf[(size_t)i * 4 + hh] - m[dst * 4 + hh]);
    ebuf[(size_t)i * 4 + hh] = ex;
    atomicAdd(&den[dst * 4 + hh], ex);
  }
}

// pass 3: agg[dst,ch] += alpha * h[src,ch]. Wave per edge; h row is L2-resident.
__global__ void edge_aggregate(const int* __restrict__ ei,
                               const float* __restrict__ h,
                               const float* __restrict__ ebuf,
                               const float* __restrict__ den,
                               float* __restrict__ agg) {
  const int lane = threadIdx.x & 31;
  const int id = blockIdx.x * 8 + (threadIdx.x >> 5);
  if (id >= ETOT) return;
  const int src = (id < NEDGES) ? ei[id] : (id - NEDGES);
  const int dst = (id < NEDGES) ? ei[NEDGES + id] : (id - NEDGES);
  const float* hs = h + (size_t)src * HC;
  float* ag = agg + (size_t)dst * HC;
#pragma unroll
  for (int j = 0; j < 8; ++j) {
    const int hh = j >> 1;  // wave-uniform head index
    const float alpha = ebuf[(size_t)id * 4 + hh] / den[dst * 4 + hh];
    const int ch = lane + j * 32;
    atomicAdd(&ag[ch], alpha * hs[ch]);
  }
}

__global__ void init_pool(float* psum, float* pmax, float* pcnt) {
  const int idx = blockIdx.x * blockDim.x + threadIdx.x;
  if (idx < NGRAPH * NCH) { psum[idx] = 0.f; pmax[idx] = -__builtin_inff(); }
  if (idx < NGRAPH) pcnt[idx] = 0.f;
}

__global__ void pool_accum(const float* __restrict__ hsel,
                           const int* __restrict__ batch,
                           float* psum, float* pmax, float* pcnt) {
  const int lane = threadIdx.x & 31;
  const int node = blockIdx.x * 8 + (threadIdx.x >> 5);
  if (node >= NNODES) return;
  const int g = batch[node];
#pragma unroll
  for (int j = 0; j < 2; ++j) {
    const int ch = lane + 32 * j;
    const float v = hsel[(size_t)node * NCH + ch];
    atomicAdd(&psum[g * NCH + ch], v);
    atomicMaxF(&pmax[g * NCH + ch], v);
  }
  if (lane == 0) atomicAdd(&pcnt[g], 1.f);
}

__global__ void pool_finalize(const float* psum, const float* pmax,
                              const float* pcnt, float* __restrict__ xout) {
  const int idx = blockIdx.x * blockDim.x + threadIdx.x;
  if (idx >= NGRAPH * NCH) return;
  const int g = idx >> 6, ch = idx & 63;
  xout[g * 128 + ch] = psum[idx] / pcnt[g];
  xout[g * 128 + 64 + ch] = pmax[idx];
}

// ---------------- head: G=64, single block, staged in LDS ----------------
__global__ void head_kernel(const float* __restrict__ x1, const float* __restrict__ x2,
                            const float* Ws, const float* bs,
                            const float* Wsh, const float* bsh,
                            const float* Wp1, const float* bp1,
                            const float* Wp2, const float* bp2,
                            const float* Wp3, const float* bp3,
                            const float* Wv1, const float* bv1,
                            const float* Wv2, const float* bv2,
                            const float* Wv3, const float* bv3,
                            const int* __restrict__ input_actions,
                            float* __restrict__ out) {
  __shared__ float bufA[64 * 128];  // 32KB: g0 / hp1+hp2 / hv1+hv2
  __shared__ float bufB[64 * 64];   // 16KB: g (post-shared-layer, kept live)
  __shared__ float lg[64 * 29];
  __shared__ float lp[5][64];
  __shared__ float en[5][64];
  __shared__ float val[64];
  const int tid = threadIdx.x, nthr = blockDim.x;

  // g0 = cat(x1,x2) @ Ws + bs   [64,128]
  for (int idx = tid; idx < 64 * 128; idx += nthr) {
    const int i = idx >> 7, j = idx & 127;
    float acc = bs[j];
    for (int k = 0; k < 128; ++k) acc += x1[i * 128 + k] * Ws[k * 128 + j];
    for (int k = 0; k < 128; ++k) acc += x2[i * 128 + k] * Ws[(128 + k) * 128 + j];
    bufA[idx] = acc;
  }
  __syncthreads();
  // g = selu(g0 @ Wsh + bsh)    [64,64] -> bufB (kept)
  for (int idx = tid; idx < 64 * 64; idx += nthr) {
    const int i = idx >> 6, j = idx & 63;
    float acc = bsh[j];
    for (int k = 0; k < 128; ++k) acc += bufA[i * 128 + k] * Wsh[k * 64 + j];
    bufB[idx] = selu_f(acc);
  }
  __syncthreads();
  // hp1 = selu(g @ Wp1 + bp1) -> bufA[0:4096]
  for (int idx = tid; idx < 64 * 64; idx += nthr) {
    const int i = idx >> 6, j = idx & 63;
    float acc = bp1[j];
    for (int k = 0; k < 64; ++k) acc += bufB[i * 64 + k] * Wp1[k * 64 + j];
    bufA[idx] = selu_f(acc);
  }
  __syncthreads();
  // hp2 = selu(hp1 @ Wp2 + bp2) -> bufA[4096:8192]
  for (int idx = tid; idx < 64 * 64; idx += nthr) {
    const int i = idx >> 6, j = idx & 63;
    float acc = bp2[j];
    for (int k = 0; k < 64; ++k) acc += bufA[i * 64 + k] * Wp2[k * 64 + j];
    bufA[4096 + idx] = selu_f(acc);
  }
  __syncthreads();
  // logits = hp2 @ Wp3 + bp3   [64,29]
  for (int idx = tid; idx < 64 * 29; idx += nthr) {
    const int i = idx / 29, j = idx % 29;
    float acc = bp3[j];
    for (int k = 0; k < 64; ++k) acc += bufA[4096 + i * 64 + k] * Wp3[k * 29 + j];
    lg[idx] = acc;
  }
  __syncthreads();
  // hv1 = selu(g @ Wv1 + bv1) -> bufA[0:4096]
  for (int idx = tid; idx < 64 * 64; idx += nthr) {
    const int i = idx >> 6, j = idx & 63;
    float acc = bv1[j];
    for (int k = 0; k < 64; ++k) acc += bufB[i * 64 + k] * Wv1[k * 64 + j];
    bufA[idx] = selu_f(acc);
  }
  __syncthreads();
  // hv2 = selu(hv1 @ Wv2 + bv2) -> bufA[4096:8192]
  for (int idx = tid; idx < 64 * 64; idx += nthr) {
    const int i = idx >> 6, j = idx & 63;
    float acc = bv2[j];
    for (int k = 0; k < 64; ++k) acc += bufA[i * 64 + k] * Wv2[k * 64 + j];
    bufA[4096 + idx] = selu_f(acc);
  }
  __syncthreads();
  // value = hv2 @ Wv3 + bv3
  for (int g = tid; g < 64; g += nthr) {
    float acc = bv3[0];
    for (int k = 0; k < 64; ++k) acc += bufA[4096 + g * 64 + k] * Wv3[k];
    val[g] = acc;
  }
  __syncthreads();
  // per-part log-softmax, chosen logp, entropy
  const int offs[5] = {0, 7, 11, 17, 21};
  const int szs[5]  = {7, 4, 6, 4, 8};
  for (int idx = tid; idx < 5 * 64; idx += nthr) {
    const int k = idx >> 6, g = idx & 63;
    const int off = offs[k], s = szs[k];
    const float* row = &lg[g * 29 + off];
    float mx = -__builtin_inff();
    for (int j = 0; j < s; ++j) mx = fmaxf(mx, row[j]);
    float se = 0.f;
    for (int j = 0; j < s; ++j) se += __expf(row[j] - mx);
    const float lse = logf(se) + mx;
    const int act = input_actions[k * 64 + g];
    lp[k][g] = row[act] - lse;
    float e = 0.f;
    for (int j = 0; j < s; ++j) {
      const float l = row[j] - lse;
      e -= __expf(l) * l;
    }
    en[k][g] = e;
  }
  __syncthreads();
  // final combination
  for (int g = tid; g < 64; g += nthr) {
    const int act = input_actions[g];  // part 0 action
    int sel = act - 1;
    sel = sel < 0 ? 0 : (sel > 4 ? 4 : sel);
    float add_lp, add_en;
    if (act < 3)       { add_lp = lp[1][g];            add_en = en[1][g]; }
    else if (act < 5)  { add_lp = lp[sel][g];          add_en = en[sel][g]; }
    else if (act == 5) { add_lp = lp[1][g] + lp[4][g]; add_en = en[1][g] + en[4][g]; }
    else               { add_lp = 0.f;                 add_en = 0.f; }
    out[g]        = lp[0][g] + add_lp;
    out[64 + g]   = en[0][g] + add_en;
    out[128 + g]  = val[g];
  }
}

// ---------------------------------------------------------------------
extern "C" void kernel_launch(void* const* d_in, const int* in_sizes, int n_in,
                              void* d_out, int out_size, void* d_ws, size_t ws_size,
                              hipStream_t stream) {
  (void)in_sizes; (void)n_in; (void)out_size; (void)ws_size;
  const float* x    = (const float*)d_in[0];
  const int*   ei   = (const int*)d_in[1];
  const int*   batch= (const int*)d_in[2];
  const int*   acts = (const int*)d_in[3];
  const float* W1   = (const float*)d_in[4];
  const float* a_s1 = (const float*)d_in[5];
  const float* a_d1 = (const float*)d_in[6];
  const float* b1   = (const float*)d_in[7];
  const float* Wl1  = (const float*)d_in[8];
  const float* bl1  = (const float*)d_in[9];
  const float* W2   = (const float*)d_in[10];
  const float* a_s2 = (const float*)d_in[11];
  const float* a_d2 = (const float*)d_in[12];
  const float* b2   = (const float*)d_in[13];
  const float* Wl2  = (const float*)d_in[14];
  const float* bl2  = (const float*)d_in[15];
  const float* Ws   = (const float*)d_in[16];
  const float* bs   = (const float*)d_in[17];
  const float* Wsh  = (const float*)d_in[18];
  const float* bsh  = (const float*)d_in[19];
  const float* Wp1  = (const float*)d_in[20];
  const float* bp1  = (const float*)d_in[21];
  const float* Wp2  = (const float*)d_in[22];
  const float* bp2  = (const float*)d_in[23];
  const float* Wp3  = (const float*)d_in[24];
  const float* bp3  = (const float*)d_in[25];
  const float* Wv1  = (const float*)d_in[26];
  const float* bv1  = (const float*)d_in[27];
  const float* Wv2  = (const float*)d_in[28];
  const float* bv2  = (const float*)d_in[29];
  const float* Wv3  = (const float*)d_in[30];
  const float* bv3  = (const float*)d_in[31];
  float* out = (float*)d_out;

  // workspace partition (floats)
  float* w = (float*)d_ws;
  size_t off = 0;
  float* hbuf = w + off; off += (size_t)NNODES * HC;     // h1 / h2
  float* agg  = w + off; off += (size_t)NNODES * HC;     // gat aggregation
  float* hsel = w + off; off += (size_t)NNODES * NCH;    // selu(linear)
  float* ebuf = w + off; off += (size_t)ETOT * NHEAD;    // edge scores / ex
  float* es   = w + off; off += (size_t)NNODES * NHEAD;
  float* ed   = w + off; off += (size_t)NNODES * NHEAD;
  float* mbuf = w + off; off += (size_t)NNODES * NHEAD;
  float* den  = w + off; off += (size_t)NNODES * NHEAD;
  float* psum = w + off; off += (size_t)NGRAPH * NCH;
  float* pmax = w + off; off += (size_t)NGRAPH * NCH;
  float* pcnt = w + off; off += (size_t)NGRAPH;
  float* x1   = w + off; off += (size_t)NGRAPH * 128;
  float* x2   = w + off; off += (size_t)NGRAPH * 128;

  const int thr = 256;
  // GEMM grids: wave computes 16x64 strip (NT=4)
  const int gemm_blk_256 = (3125 * 4 + 7) / 8;   // NCOLS=256 -> ntn=4
  const int gemm_blk_64  = (3125 * 1 + 7) / 8;   // NCOLS=64  -> ntn=1
  const int attn_blk = (NNODES * NHEAD + 7) / 8;
  const int initL_blk = (NNODES * HC + thr - 1) / thr;
  const int edge_blk = (ETOT + thr - 1) / thr;
  const int eagg_blk = (ETOT + 7) / 8;
  const int poolI_blk = (NGRAPH * NCH + thr - 1) / thr;
  const int poolA_blk = (NNODES + 7) / 8;

  // ---- layer 1 ----
  wmma_gemm_t<NIN, NIN, HC, HC, 4, false, false, false>
      <<<gemm_blk_256, thr, 0, stream>>>(x, nullptr, W1, hbuf, nullptr, NNODES);
  attn_scores<<<attn_blk, thr, 0, stream>>>(hbuf, a_s1, a_d1, es, ed);
  init_layer<<<initL_blk, thr, 0, stream>>>(agg, mbuf, den);
  edge_scores<<<edge_blk, thr, 0, stream>>>(ei, es, ed, ebuf, mbuf);
  edge_softmax_num<<<edge_blk, thr, 0, stream>>>(ei, mbuf, ebuf, den);
  edge_aggregate<<<eagg_blk, thr, 0, stream>>>(ei, hbuf, ebuf, den, agg);
  wmma_gemm_t<HC, HC, NCH, NCH, 4, true, true, true>
      <<<gemm_blk_64, thr, 0, stream>>>(agg, b1, Wl1, hsel, bl1, NNODES);
  init_pool<<<poolI_blk, thr, 0, stream>>>(psum, pmax, pcnt);
  pool_accum<<<poolA_blk, thr, 0, stream>>>(hsel, batch, psum, pmax, pcnt);
  pool_finalize<<<poolI_blk, thr, 0, stream>>>(psum, pmax, pcnt, x1);

  // ---- layer 2 ----
  wmma_gemm_t<NCH, NCH, HC, HC, 4, false, false, false>
      <<<gemm_blk_256, thr, 0, stream>>>(hsel, nullptr, W2, hbuf, nullptr, NNODES);
  attn_scores<<<attn_blk, thr, 0, stream>>>(hbuf, a_s2, a_d2, es, ed);
  init_layer<<<initL_blk, thr, 0, stream>>>(agg, mbuf, den);
  edge_scores<<<edge_blk, thr, 0, stream>>>(ei, es, ed, ebuf, mbuf);
  edge_softmax_num<<<edge_blk, thr, 0, stream>>>(ei, mbuf, ebuf, den);
  edge_aggregate<<<eagg_blk, thr, 0, stream>>>(ei, hbuf, ebuf, den, agg);
  wmma_gemm_t<HC, HC, NCH, NCH, 4, true, true, true>
      <<<gemm_blk_64, thr, 0, stream>>>(agg, b2, Wl2, hsel, bl2, NNODES);
  init_pool<<<poolI_blk, thr, 0, stream>>>(psum, pmax, pcnt);
  pool_accum<<<poolA_blk, thr, 0, stream>>>(hsel, batch, psum, pmax, pcnt);
  pool_finalize<<<poolI_blk, thr, 0, stream>>>(psum, pmax, pcnt, x2);

  // ---- head ----
  head_kernel<<<1, 256, 0, stream>>>(x1, x2, Ws, bs, Wsh, bsh,
                                     Wp1, bp1, Wp2, bp2, Wp3, bp3,
                                     Wv1, bv1, Wv2, bv2, Wv3, bv3,
                                     acts, out);
}